// CausalGCN_20968030339554
// MI455X (gfx1250) — compile-verified
//
#include <hip/hip_runtime.h>

// ---------------------------------------------------------------------------
// CausalGCN forward for MI455X (gfx1250, wave32, WMMA).
// Dense 128-K GEMMs on v_wmma_f32_16x16x4_f32 (full f32 fidelity; workload is
// L2/scatter-bound so low-K f32 WMMA is not the bottleneck). BatchNorm is
// folded into the weights once per layer, so X tiles are staged raw into LDS
// via CDNA5 async global->LDS (ASYNCcnt) when the builtin is available.
// ---------------------------------------------------------------------------

#define N_NODES 50000
#define N_EDGES 600000
#define HDIM    128
#define NCLS    10
#define NGRAPH  256
#define NLAYER  3

#define BN_PARTS  240
#define GEMM_ROWS 64
#define XS_STRIDE 132   // 132*4 B row stride: 16B-aligned rows, conflict-free A reads

#if defined(__has_builtin)
#if __has_builtin(__builtin_amdgcn_global_load_async_to_lds_b128)
#define HAVE_ASYNC_LDS 1
#endif
#endif
#ifndef HAVE_ASYNC_LDS
#define HAVE_ASYNC_LDS 0
#endif

typedef float v2f __attribute__((ext_vector_type(2)));
typedef float v8f __attribute__((ext_vector_type(8)));
typedef int   v4i __attribute__((vector_size(16)));   // matches builtin's param type

__device__ __forceinline__ void wait_async_lds() {
#if HAVE_ASYNC_LDS
#if defined(__has_builtin) && __has_builtin(__builtin_amdgcn_s_wait_asynccnt)
  __builtin_amdgcn_s_wait_asynccnt(0);
#else
  asm volatile("s_wait_asynccnt 0x0" ::: "memory");
#endif
#endif
}

__device__ __forceinline__ void stage_chunk(const float* gp, float* lp) {
#if HAVE_ASYNC_LDS
  __builtin_amdgcn_global_load_async_to_lds_b128((v4i*)gp, (v4i*)lp, 0, 0);
#else
  *(float4*)lp = *(const float4*)gp;
#endif
}

// ---------------- elementwise helpers ----------------
__global__ void k_zero(float* __restrict__ p, int n) {
  int i = blockIdx.x * blockDim.x + threadIdx.x;
  if (i < n) p[i] = 0.f;
}

__global__ void k_relu(float* __restrict__ p, int n) {
  int i = blockIdx.x * blockDim.x + threadIdx.x;
  if (i < n) p[i] = fmaxf(p[i], 0.f);
}

// ---------------- BatchNorm statistics (deterministic 2-pass) --------------
// blockDim.x == Fdim (128 or 256); each block accumulates a row-stripe.
__global__ void k_bn_partial(const float* __restrict__ X, int n, int Fdim,
                             float* __restrict__ psum, float* __restrict__ psq) {
  int f = threadIdx.x;
  float s = 0.f, s2 = 0.f;
  for (int r = blockIdx.x; r < n; r += gridDim.x) {
    float v = X[(size_t)r * Fdim + f];
    s += v;
    s2 += v * v;
  }
  psum[(size_t)blockIdx.x * Fdim + f] = s;
  psq [(size_t)blockIdx.x * Fdim + f] = s2;
}

// single block, blockDim.x == Fdim. Produces fused affine scale/shift.
__global__ void k_bn_fin(const float* __restrict__ psum, const float* __restrict__ psq,
                         int nparts, int Fdim, int n,
                         const float* __restrict__ g, const float* __restrict__ b,
                         float* __restrict__ scl, float* __restrict__ shf) {
  int f = threadIdx.x;
  float s = 0.f, s2 = 0.f;
  for (int p = 0; p < nparts; ++p) {
    s  += psum[(size_t)p * Fdim + f];
    s2 += psq [(size_t)p * Fdim + f];
  }
  float mean = s / (float)n;
  float var  = s2 / (float)n - mean * mean;   // biased var, matches jnp.var
  float inv  = rsqrtf(var + 1e-5f);
  float sc   = g[f] * inv;
  scl[f] = sc;
  shf[f] = b[f] - mean * sc;
}

// ---------------- fold BN affine into weights --------------------------
// Wf[k][j] = scl[k]*W[k][j];  bf[j] = (bias?bias[j]:0) + sum_k shf[k]*W[k][j]
// so  bn(X) @ W + bias  ==  X @ Wf + bf.   Single block, 128 threads.
__global__ __launch_bounds__(128) void k_fold_bn(
    const float* __restrict__ scl, const float* __restrict__ shf,
    const float* __restrict__ W, const float* __restrict__ bias, int K,
    float* __restrict__ Wf, float* __restrict__ bf) {
  int j = threadIdx.x;
  float acc = bias ? bias[j] : 0.f;
  for (int k = 0; k < K; ++k) {
    float wkj = W[(size_t)k * 128 + j];
    Wf[(size_t)k * 128 + j] = scl[k] * wkj;
    acc += shf[k] * wkj;
  }
  bf[j] = acc;
}

// ---------------- WMMA GEMM: Y[n,128] = X[n,128] @ W[128,128] + b ----------
// 8 waves/block, each wave owns a 16-col strip; block covers 64 rows.
// A-layout (16x4 f32): lane<16 -> (M=lane, K=kb..kb+1), lane>=16 -> K=kb+2..kb+3
// C/D-layout (16x16 f32): VGPR v, lanes<16 -> M=v, lanes>=16 -> M=v+8, N=lane%16
__global__ __launch_bounds__(256) void k_gemm128(
    const float* __restrict__ X, int nrows,
    const float* __restrict__ W,        // folded [128][128] row-major (k, j)
    const float* __restrict__ bias,     // folded [128]
    float* __restrict__ Y, int doRelu) {
  __shared__ float Xs[GEMM_ROWS][XS_STRIDE];
  const int tid = threadIdx.x;
  const int rowBase = blockIdx.x * GEMM_ROWS;
  const bool fullTile = (rowBase + GEMM_ROWS) <= nrows;

  // stage raw 64x128 X tile in 16B chunks (async global->LDS on CDNA5)
  if (fullTile) {
#pragma unroll
    for (int i = 0; i < 8; ++i) {            // 2048 chunks / 256 threads
      int idx = tid + i * 256;
      int r   = idx >> 5;                    // 32 chunks per row
      int c4  = (idx & 31) * 4;
      stage_chunk(X + (size_t)(rowBase + r) * HDIM + c4, &Xs[r][c4]);
    }
  } else {
    for (int i = tid; i < GEMM_ROWS * (HDIM / 4); i += 256) {
      int r  = i >> 5;
      int c4 = (i & 31) * 4;
      int gr = rowBase + r;
      float* lp = &Xs[r][c4];
      if (gr < nrows) {
        stage_chunk(X + (size_t)gr * HDIM + c4, lp);
      } else {
        lp[0] = 0.f; lp[1] = 0.f; lp[2] = 0.f; lp[3] = 0.f;
      }
    }
  }
  wait_async_lds();
  __syncthreads();

  const int lane = tid & 31;
  const int half = lane >> 4;   // 0: K pair {0,1}, 1: K pair {2,3}
  const int lr   = lane & 15;
  const int col  = (tid >> 5) * 16 + lr;
  const float bcol = bias[col];

  v8f acc[GEMM_ROWS / 16];
  const v8f vz = {0.f, 0.f, 0.f, 0.f, 0.f, 0.f, 0.f, 0.f};
#pragma unroll
  for (int rt = 0; rt < GEMM_ROWS / 16; ++rt) acc[rt] = vz;

  for (int ks = 0; ks < 32; ++ks) {           // K = 128 in steps of 4
    const int kb = ks * 4 + half * 2;
    v2f bmat;
    bmat.x = W[(size_t)kb * HDIM + col];
    bmat.y = W[(size_t)(kb + 1) * HDIM + col];
#pragma unroll
    for (int rt = 0; rt < GEMM_ROWS / 16; ++rt) {
      v2f amat;
      amat.x = Xs[rt * 16 + lr][kb];
      amat.y = Xs[rt * 16 + lr][kb + 1];
      acc[rt] = __builtin_amdgcn_wmma_f32_16x16x4_f32(
          false, amat, false, bmat, (short)0, acc[rt], false, false);
    }
  }

  // epilogue: straight-line stores for full tiles (common case), guarded tail
  if (fullTile) {
    float* yb = Y + (size_t)(rowBase + half * 8) * HDIM + col;
    if (doRelu) {
#pragma unroll
      for (int rt = 0; rt < GEMM_ROWS / 16; ++rt)
#pragma unroll
        for (int v = 0; v < 8; ++v)
          yb[(size_t)(rt * 16 + v) * HDIM] = fmaxf(acc[rt][v] + bcol, 0.f);
    } else {
#pragma unroll
      for (int rt = 0; rt < GEMM_ROWS / 16; ++rt)
#pragma unroll
        for (int v = 0; v < 8; ++v)
          yb[(size_t)(rt * 16 + v) * HDIM] = acc[rt][v] + bcol;
    }
  } else {
#pragma unroll
    for (int rt = 0; rt < GEMM_ROWS / 16; ++rt)
#pragma unroll
      for (int v = 0; v < 8; ++v) {
        int gr = rowBase + rt * 16 + v + half * 8;
        if (gr < nrows) {
          float y = acc[rt][v] + bcol;
          if (doRelu) y = fmaxf(y, 0.f);
          Y[(size_t)gr * HDIM + col] = y;
        }
      }
  }
}

// ---------------- GCN aggregation ----------------
// out[n,:] = dinv[n]^2 * xw[n,:] + bias  (self loop, weight 1)
__global__ void k_selfloop(const float* __restrict__ xw, const float* __restrict__ dinv,
                           const float* __restrict__ bias, float* __restrict__ out, int n) {
  int i = blockIdx.x * blockDim.x + threadIdx.x;
  if (i < n * HDIM) {
    int r = i >> 7, f = i & 127;
    float d = dinv[r];
    out[i] = d * d * xw[i] + bias[f];
  }
}

// out[row[e],:] += dinv[row]*dinv[col]*ew[e] * xw[col[e],:]
__global__ __launch_bounds__(256) void k_edge_scatter(
    const int* __restrict__ row, const int* __restrict__ colx,
    const float* __restrict__ ew, const float* __restrict__ dinv,
    const float* __restrict__ xw, float* __restrict__ out, int nE) {
  int e = blockIdx.x * 2 + (threadIdx.x >> 7);
  int f = threadIdx.x & 127;
  if (e < nE) {
    int r = row[e], c = colx[e];
    float wgt = dinv[r] * dinv[c] * (ew ? ew[e] : 1.f);
    atomicAdd(&out[(size_t)r * HDIM + f], wgt * xw[(size_t)c * HDIM + f]);
  }
}

__global__ void k_deg_count(const int* __restrict__ row, const float* __restrict__ ew,
                            float* __restrict__ deg, int nE) {
  int e = blockIdx.x * blockDim.x + threadIdx.x;
  if (e < nE) atomicAdd(&deg[row[e]], ew ? ew[e] : 1.f);
}

__global__ void k_deg_fin(float* __restrict__ deg, int n) {
  int i = blockIdx.x * blockDim.x + threadIdx.x;
  if (i < n) deg[i] = rsqrtf(deg[i] + 1.f);   // +1 self loop; always > 0
}

// ---------------- attention (one wave per edge / node) ----------------
__global__ __launch_bounds__(256) void k_edge_att(
    const int* __restrict__ row, const int* __restrict__ colx, const float* __restrict__ h,
    const float* __restrict__ Wc, const float* __restrict__ bc,
    const float* __restrict__ Wo, const float* __restrict__ bo,
    float* __restrict__ ewc, float* __restrict__ ewo, int nE) {
  int lane = threadIdx.x & 31;
  int e = blockIdx.x * 8 + (threadIdx.x >> 5);
  if (e >= nE) return;
  int r = row[e], c = colx[e];
  float ac = 0.f, ao = 0.f;
#pragma unroll
  for (int q = 0; q < 4; ++q) {
    int f = lane + q * 32;
    float hr = h[(size_t)r * HDIM + f];
    float hc = h[(size_t)c * HDIM + f];
    ac += hr * Wc[f] + hc * Wc[HDIM + f];
    ao += hr * Wo[f] + hc * Wo[HDIM + f];
  }
#pragma unroll
  for (int s = 16; s > 0; s >>= 1) {
    ac += __shfl_xor(ac, s, 32);
    ao += __shfl_xor(ao, s, 32);
  }
  if (lane == 0) {
    float lc = ac + bc[0], lo = ao + bo[0];
    float m = fmaxf(lc, lo);
    float ec = __expf(lc - m), eo = __expf(lo - m);
    float inv = 1.f / (ec + eo);
    ewc[e] = ec * inv;
    ewo[e] = eo * inv;
  }
}

__global__ __launch_bounds__(256) void k_node_att(
    const float* __restrict__ h,
    const float* __restrict__ Wc, const float* __restrict__ bc,
    const float* __restrict__ Wo, const float* __restrict__ bo,
    float* __restrict__ xc, float* __restrict__ xo, int n) {
  int lane = threadIdx.x & 31;
  int nd = blockIdx.x * 8 + (threadIdx.x >> 5);
  if (nd >= n) return;
  float hv[4];
  float ac = 0.f, ao = 0.f;
#pragma unroll
  for (int q = 0; q < 4; ++q) {
    int f = lane + q * 32;
    hv[q] = h[(size_t)nd * HDIM + f];
    ac += hv[q] * Wc[f];
    ao += hv[q] * Wo[f];
  }
#pragma unroll
  for (int s = 16; s > 0; s >>= 1) {     // butterfly: all lanes get full sum
    ac += __shfl_xor(ac, s, 32);
    ao += __shfl_xor(ao, s, 32);
  }
  float lc = ac + bc[0], lo = ao + bo[0];
  float m = fmaxf(lc, lo);
  float ec = __expf(lc - m), eo = __expf(lo - m);
  float inv = 1.f / (ec + eo);
  float wc = ec * inv, wo = eo * inv;
#pragma unroll
  for (int q = 0; q < 4; ++q) {
    int f = lane + q * 32;
    xc[(size_t)nd * HDIM + f] = wc * hv[q];
    xo[(size_t)nd * HDIM + f] = wo * hv[q];
  }
}

// ---------------- pooling ----------------
__global__ void k_pool(const float* __restrict__ x, const int* __restrict__ batch,
                       float* __restrict__ pool, int n) {
  int i = blockIdx.x * blockDim.x + threadIdx.x;
  if (i < n * HDIM) {
    int r = i >> 7, f = i & 127;
    atomicAdd(&pool[(size_t)batch[r] * HDIM + f], x[i]);
  }
}

__global__ void k_concat(const float* __restrict__ a, const float* __restrict__ b,
                         float* __restrict__ o, int nrows) {
  int i = blockIdx.x * blockDim.x + threadIdx.x;
  if (i < nrows * 256) {
    int r = i >> 8, f = i & 255;
    o[i] = (f < 128) ? a[(size_t)r * 128 + f] : b[(size_t)r * 128 + (f - 128)];
  }
}

// ---------------- small readout kernels ----------------
// Y[g,128] = relu(X[g,256] @ Wf[256,128] + bf)   (co1, BN already folded)
__global__ __launch_bounds__(128) void k_gemm_k256(
    const float* __restrict__ X, const float* __restrict__ W,
    const float* __restrict__ bias, float* __restrict__ Y, int nrows) {
  __shared__ float xr[256];
  int gI = blockIdx.x, j = threadIdx.x;
  if (gI >= nrows) return;
  xr[j]       = X[(size_t)gI * 256 + j];
  xr[j + 128] = X[(size_t)gI * 256 + j + 128];
  __syncthreads();
  float acc = bias[j];
  for (int k = 0; k < 256; ++k) acc += xr[k] * W[(size_t)k * 128 + j];
  Y[(size_t)gI * 128 + j] = fmaxf(acc, 0.f);
}

// out[g,0..9] = log_softmax(bn(Z[g,:]) @ W[128,10] + bias)
__global__ __launch_bounds__(128) void k_proj_lsm(
    const float* __restrict__ Z, const float* __restrict__ scl, const float* __restrict__ shf,
    const float* __restrict__ W, const float* __restrict__ bias, float* __restrict__ out) {
  __shared__ float zr[128];
  __shared__ float logit[NCLS];
  int gI = blockIdx.x, t = threadIdx.x;
  zr[t] = Z[(size_t)gI * 128 + t] * scl[t] + shf[t];
  __syncthreads();
  if (t < NCLS) {
    float a = bias[t];
    for (int k = 0; k < 128; ++k) a += zr[k] * W[(size_t)k * NCLS + t];
    logit[t] = a;
  }
  __syncthreads();
  if (t == 0) {
    float m = -1e30f;
    for (int c = 0; c < NCLS; ++c) m = fmaxf(m, logit[c]);
    float s = 0.f;
    for (int c = 0; c < NCLS; ++c) s += __expf(logit[c] - m);
    float lse = m + __logf(s);
    for (int c = 0; c < NCLS; ++c) out[(size_t)gI * NCLS + c] = logit[c] - lse;
  }
}

// ---------------------------------------------------------------------------
extern "C" void kernel_launch(void* const* d_in, const int* in_sizes, int n_in,
                              void* d_out, int out_size, void* d_ws, size_t ws_size,
                              hipStream_t stream) {
  (void)in_sizes; (void)n_in; (void)out_size; (void)ws_size;
  const int N = N_NODES, E = N_EDGES;

  const float* x    = (const float*)d_in[0];
  const int* ei     = (const int*)d_in[1];
  const int* batch  = (const int*)d_in[2];
  const int* row    = ei;          // edge_index[0] (targets)
  const int* col    = ei + E;      // edge_index[1] (sources)
  auto P = [&](int i) { return (const float*)d_in[i]; };

  // ---- workspace carve-up (floats), ~83 MB total ----
  const size_t NH = (size_t)N * HDIM;
  float* w = (float*)d_ws;
  float* bufA  = w; w += NH;
  float* bufB  = w; w += NH;
  float* bufD  = w; w += NH;
  float* dinv  = w; w += N;                 // unweighted-conv 1/sqrt(deg)
  float* dinvA = w; w += N;                 // attention-conv 1/sqrt(deg)
  float* ewc   = w; w += E;
  float* ewo   = w; w += E;
  float* scl   = w; w += 256;
  float* shf   = w; w += 256;
  float* psum  = w; w += (size_t)BN_PARTS * 256;
  float* psq   = w; w += (size_t)BN_PARTS * 256;
  float* poolc = w; w += (size_t)NGRAPH * HDIM;
  float* poolo = w; w += (size_t)NGRAPH * HDIM;
  float* cat   = w; w += (size_t)NGRAPH * 256;
  float* z1    = w; w += (size_t)NGRAPH * HDIM;
  float* Wf    = w; w += (size_t)256 * 128;   // folded weights (max K=256)
  float* bf    = w; w += 128;                 // folded bias

  dim3 B256(256), B128(128);
  const int gElem = (int)((NH + 255) / 256);
  const int gN    = (N + 255) / 256;
  const int gE    = (E + 255) / 256;
  const int gGemm = (N + GEMM_ROWS - 1) / GEMM_ROWS;
  const int gScat = (E + 1) / 2;
  const int gEAtt = (E + 7) / 8;
  const int gNAtt = (N + 7) / 8;
  const int gPoolZ = (NGRAPH * HDIM + 255) / 256;
  float* out = (float*)d_out;

  // ---- unweighted symmetric degree (shared by the 3 mid convs) ----
  k_zero<<<gN, B256, 0, stream>>>(dinv, N);
  k_deg_count<<<gE, B256, 0, stream>>>(row, nullptr, dinv, E);
  k_deg_fin<<<gN, B256, 0, stream>>>(dinv, N);

  // ---- feature BN + gfn conv: h = relu(bn(x) @ Wf + bf) ----
  k_bn_partial<<<BN_PARTS, B128, 0, stream>>>(x, N, 128, psum, psq);
  k_bn_fin<<<1, B128, 0, stream>>>(psum, psq, BN_PARTS, 128, N, P(3), P(4), scl, shf);
  k_fold_bn<<<1, B128, 0, stream>>>(scl, shf, P(5), P(6), 128, Wf, bf);
  k_gemm128<<<gGemm, B256, 0, stream>>>(x, N, Wf, bf, bufA, 1);

  // ---- 3 GCN layers ----
  float* h = bufA; float* t = bufB; float* o = bufD;
  for (int i = 0; i < NLAYER; ++i) {
    k_bn_partial<<<BN_PARTS, B128, 0, stream>>>(h, N, 128, psum, psq);
    k_bn_fin<<<1, B128, 0, stream>>>(psum, psq, BN_PARTS, 128, N,
                                     P(7) + i * HDIM, P(8) + i * HDIM, scl, shf);
    k_fold_bn<<<1, B128, 0, stream>>>(scl, shf, P(9) + (size_t)i * HDIM * HDIM,
                                      nullptr, 128, Wf, bf);
    k_gemm128<<<gGemm, B256, 0, stream>>>(h, N, Wf, bf, t, 0);
    k_selfloop<<<gElem, B256, 0, stream>>>(t, dinv, P(10) + i * HDIM, o, N);
    k_edge_scatter<<<gScat, B256, 0, stream>>>(row, col, nullptr, dinv, t, o, E);
    k_relu<<<gElem, B256, 0, stream>>>(o, (int)NH);
    float* tmp = h; h = o; o = tmp;
  }

  // ---- edge & node attention ----
  k_edge_att<<<gEAtt, B256, 0, stream>>>(row, col, h, P(11), P(12), P(13), P(14), ewc, ewo, E);
  k_node_att<<<gNAtt, B256, 0, stream>>>(h, P(15), P(16), P(17), P(18), t, o, N);
  // now t = xc_att, o = xo_att; h buffer is free

  // ---- context branch ----
  k_bn_partial<<<BN_PARTS, B128, 0, stream>>>(t, N, 128, psum, psq);
  k_bn_fin<<<1, B128, 0, stream>>>(psum, psq, BN_PARTS, 128, N, P(19), P(20), scl, shf);
  k_fold_bn<<<1, B128, 0, stream>>>(scl, shf, P(23), nullptr, 128, Wf, bf);
  k_gemm128<<<gGemm, B256, 0, stream>>>(t, N, Wf, bf, h, 0);
  k_zero<<<gN, B256, 0, stream>>>(dinvA, N);
  k_deg_count<<<gE, B256, 0, stream>>>(row, ewc, dinvA, E);
  k_deg_fin<<<gN, B256, 0, stream>>>(dinvA, N);
  k_selfloop<<<gElem, B256, 0, stream>>>(h, dinvA, P(24), t, N);
  k_edge_scatter<<<gScat, B256, 0, stream>>>(row, col, ewc, dinvA, h, t, E);
  k_relu<<<gElem, B256, 0, stream>>>(t, (int)NH);
  k_zero<<<gPoolZ, B256, 0, stream>>>(poolc, NGRAPH * HDIM);
  k_pool<<<gElem, B256, 0, stream>>>(t, batch, poolc, N);

  // ---- object branch ----
  k_bn_partial<<<BN_PARTS, B128, 0, stream>>>(o, N, 128, psum, psq);
  k_bn_fin<<<1, B128, 0, stream>>>(psum, psq, BN_PARTS, 128, N, P(21), P(22), scl, shf);
  k_fold_bn<<<1, B128, 0, stream>>>(scl, shf, P(25), nullptr, 128, Wf, bf);
  k_gemm128<<<gGemm, B256, 0, stream>>>(o, N, Wf, bf, h, 0);
  k_zero<<<gN, B256, 0, stream>>>(dinvA, N);
  k_deg_count<<<gE, B256, 0, stream>>>(row, ewo, dinvA, E);
  k_deg_fin<<<gN, B256, 0, stream>>>(dinvA, N);
  k_selfloop<<<gElem, B256, 0, stream>>>(h, dinvA, P(26), o, N);
  k_edge_scatter<<<gScat, B256, 0, stream>>>(row, col, ewo, dinvA, h, o, E);
  k_relu<<<gElem, B256, 0, stream>>>(o, (int)NH);
  k_zero<<<gPoolZ, B256, 0, stream>>>(poolo, NGRAPH * HDIM);
  k_pool<<<gElem, B256, 0, stream>>>(o, batch, poolo, N);

  const int gGemmG = (NGRAPH + GEMM_ROWS - 1) / GEMM_ROWS;

  // ---- readout c ----
  k_bn_partial<<<32, B128, 0, stream>>>(poolc, NGRAPH, 128, psum, psq);
  k_bn_fin<<<1, B128, 0, stream>>>(psum, psq, 32, 128, NGRAPH, P(27), P(28), scl, shf);
  k_fold_bn<<<1, B128, 0, stream>>>(scl, shf, P(29), P(30), 128, Wf, bf);
  k_gemm128<<<gGemmG, B256, 0, stream>>>(poolc, NGRAPH, Wf, bf, z1, 1);
  k_bn_partial<<<32, B128, 0, stream>>>(z1, NGRAPH, 128, psum, psq);
  k_bn_fin<<<1, B128, 0, stream>>>(psum, psq, 32, 128, NGRAPH, P(31), P(32), scl, shf);
  k_proj_lsm<<<NGRAPH, B128, 0, stream>>>(z1, scl, shf, P(33), P(34), out);

  // ---- readout o ----
  k_bn_partial<<<32, B128, 0, stream>>>(poolo, NGRAPH, 128, psum, psq);
  k_bn_fin<<<1, B128, 0, stream>>>(psum, psq, 32, 128, NGRAPH, P(35), P(36), scl, shf);
  k_fold_bn<<<1, B128, 0, stream>>>(scl, shf, P(37), P(38), 128, Wf, bf);
  k_gemm128<<<gGemmG, B256, 0, stream>>>(poolo, NGRAPH, Wf, bf, z1, 1);
  k_bn_partial<<<32, B128, 0, stream>>>(z1, NGRAPH, 128, psum, psq);
  k_bn_fin<<<1, B128, 0, stream>>>(psum, psq, 32, 128, NGRAPH, P(39), P(40), scl, shf);
  k_proj_lsm<<<NGRAPH, B128, 0, stream>>>(z1, scl, shf, P(41), P(42), out + NGRAPH * NCLS);

  // ---- readout co ----
  k_concat<<<(NGRAPH * 256 + 255) / 256, B256, 0, stream>>>(poolc, poolo, cat, NGRAPH);
  k_bn_partial<<<32, B256, 0, stream>>>(cat, NGRAPH, 256, psum, psq);
  k_bn_fin<<<1, B256, 0, stream>>>(psum, psq, 32, 256, NGRAPH, P(43), P(44), scl, shf);
  k_fold_bn<<<1, B128, 0, stream>>>(scl, shf, P(45), P(46), 256, Wf, bf);
  k_gemm_k256<<<NGRAPH, B128, 0, stream>>>(cat, Wf, bf, z1, NGRAPH);
  k_bn_partial<<<32, B128, 0, stream>>>(z1, NGRAPH, 128, psum, psq);
  k_bn_fin<<<1, B128, 0, stream>>>(psum, psq, 32, 128, NGRAPH, P(47), P(48), scl, shf);
  k_proj_lsm<<<NGRAPH, B128, 0, stream>>>(z1, scl, shf, P(49), P(50), out + 2 * NGRAPH * NCLS);
}